// HardNegativeMining_549755814091
// MI455X (gfx1250) — compile-verified
//
#include <hip/hip_runtime.h>
#include <stdint.h>

// ---------------- problem constants (baked from reference) ----------------
#define BQ    2048
#define NQ    100000
#define KOUT  128          // min(num_hard_negatives+1, N)
#define KSEL  127          // negatives selected (positive handled separately)
#define TPB   256          // 8 wave32 per workgroup

// async streaming: each chunk = TPB lanes * 16B = 4 KB = 1024 floats
#define CHUNK      1024
#define NFULL      (NQ / CHUNK)        // 97 full chunks
#define TAIL_START (NFULL * CHUNK)     // 99328

// ---------------- LDS layout (dynamic block starts at LDS offset 0) -------
#define HIST_BINS 65536
#define HIST_OFF  0                         // 65536 * 4 = 262144 B
#define STAGE_OFF 262144                    // 2 * 4096 B double buffer
#define CAND_OFF  (STAGE_OFF + 8192)        // 128 * 4 B
#define BSUM_OFF  (CAND_OFF + 512)          // 256 * 4 B
#define SSUM_OFF  (BSUM_OFF + 1024)         // 256 * 4 B (scan workspace)
#define SCAL_OFF  (SSUM_OFF + 1024)         // scalars
#define SMEM_BYTES (SCAL_OFF + 64)          // 272 960 B  (< 320 KB WGP LDS)

// native vector type usable with __builtin_nontemporal_load
typedef float vfloat4 __attribute__((ext_vector_type(4)));

// order-preserving float->uint key: bigger float => bigger key
__device__ __forceinline__ uint32_t f2key(float f) {
    uint32_t u = __float_as_uint(f);
    return (u & 0x80000000u) ? ~u : (u | 0x80000000u);
}

// inclusive suffix sum over 256-entry LDS array (Hillis-Steele, 8 steps)
__device__ __forceinline__ void suffix_scan256(uint32_t* a) {
    const int tid = threadIdx.x;
    #pragma unroll
    for (int off = 1; off < 256; off <<= 1) {
        __syncthreads();
        uint32_t v = a[tid] + ((tid + off < 256) ? a[tid + off] : 0u);
        __syncthreads();
        a[tid] = v;
    }
    __syncthreads();
}

// Find bin `sel` (scanning from the top) such that
// count(bin > sel) < target <= count(bin >= sel).  Fully parallel.
// scal[1] = sel, scal[2] = count strictly greater than sel.
__device__ void suffix_select(uint32_t* hist, uint32_t* bsum, uint32_t* ssum,
                              uint32_t* scal, uint32_t target) {
    const int tid = threadIdx.x;
    // level 1: 256-bin blocks
    uint32_t s = 0;
    const uint32_t* base = hist + tid * 256;
    #pragma unroll 8
    for (int b = 0; b < 256; ++b) s += base[b];
    bsum[tid] = s;
    ssum[tid] = s;
    suffix_scan256(ssum);
    {
        uint32_t excl = (tid < 255) ? ssum[tid + 1] : 0u;   // count above my block
        if (excl < target && excl + s >= target) {
            scal[5] = (uint32_t)tid;
            scal[6] = excl;
        }
    }
    __syncthreads();
    const uint32_t blk = scal[5];
    const uint32_t acc = scal[6];
    __syncthreads();
    // level 2: bins inside selected block
    uint32_t c = hist[blk * 256 + tid];
    ssum[tid] = c;
    suffix_scan256(ssum);
    {
        uint32_t excl = acc + ((tid < 255) ? ssum[tid + 1] : 0u);
        if (excl < target && excl + c >= target) {
            scal[1] = blk * 256 + (uint32_t)tid;
            scal[2] = excl;
        }
    }
    __syncthreads();
}

extern "C" __global__ __launch_bounds__(TPB)
void hnm_topk_kernel(const float* __restrict__ logits,
                     const float* __restrict__ labels,
                     float* __restrict__ out_logits,
                     float* __restrict__ out_labels) {
    extern __shared__ char smem[];
    uint32_t* hist = (uint32_t*)(smem + HIST_OFF);
    float*    cand = (float*)(smem + CAND_OFF);
    uint32_t* bsum = (uint32_t*)(smem + BSUM_OFF);
    uint32_t* ssum = (uint32_t*)(smem + SSUM_OFF);
    uint32_t* scal = (uint32_t*)(smem + SCAL_OFF);

    const int row = blockIdx.x;
    const int tid = threadIdx.x;
    const float* lrow = logits + (size_t)row * NQ;
    const float* brow = labels + (size_t)row * NQ;

    // vectorized histogram clear (b128 LDS stores)
    {
        uint4* h4 = (uint4*)hist;
        const uint4 z = make_uint4(0u, 0u, 0u, 0u);
        for (int i = tid; i < HIST_BINS / 4; i += TPB) h4[i] = z;
    }
    if (tid < 16) scal[tid] = 0u;
    __syncthreads();

    // ---- pass 0: find positive index p (labels one-hot, NT: no L2 reuse) ----
    {
        const vfloat4* b4 = (const vfloat4*)brow;         // NQ % 4 == 0
        for (int i = tid; i < NQ / 4; i += TPB) {
            vfloat4 v = __builtin_nontemporal_load(&b4[i]);
            if (v.x != 0.f) scal[0] = 4u * i + 0u;
            if (v.y != 0.f) scal[0] = 4u * i + 1u;
            if (v.z != 0.f) scal[0] = 4u * i + 2u;
            if (v.w != 0.f) scal[0] = 4u * i + 3u;
        }
    }
    __syncthreads();
    const uint32_t p = scal[0];

    // ---- pass 1: histogram of key>>16, streamed HBM->LDS with async b128 ----
    {
        const uint64_t ga0  = (uint64_t)(uintptr_t)lrow + (uint64_t)tid * 16u;
        // prime chunk 0 into buffer 0
        {
            uint32_t la = (uint32_t)(STAGE_OFF + tid * 16);
            uint64_t ga = ga0;
            asm volatile("global_load_async_to_lds_b128 %0, %1, off"
                         :: "v"(la), "v"(ga) : "memory");
        }
        for (int c = 0; c < NFULL; ++c) {
            const uint32_t cur = (uint32_t)(STAGE_OFF + (c & 1) * 4096 + tid * 16);
            if (c + 1 < NFULL) {
                uint32_t la = (uint32_t)(STAGE_OFF + ((c + 1) & 1) * 4096 + tid * 16);
                uint64_t ga = ga0 + (uint64_t)(c + 1) * (CHUNK * 4);
                asm volatile("global_load_async_to_lds_b128 %0, %1, off"
                             :: "v"(la), "v"(ga) : "memory");
                asm volatile("s_wait_asynccnt 0x1" ::: "memory"); // current chunk done
            } else {
                asm volatile("s_wait_asynccnt 0x0" ::: "memory");
            }
            const float4 v = *(const float4*)(smem + cur);  // each lane reads its own 16B
            const uint32_t base = (uint32_t)(c * CHUNK + tid * 4);
            #pragma unroll
            for (int e = 0; e < 4; ++e) {
                uint32_t idx = base + (uint32_t)e;
                if (idx != p) {
                    uint32_t key = f2key((&v.x)[e]);
                    atomicAdd(&hist[key >> 16], 1u);
                }
            }
        }
        for (int i = TAIL_START + tid; i < NQ; i += TPB) {   // 672 tail elements
            if ((uint32_t)i != p) atomicAdd(&hist[f2key(lrow[i]) >> 16], 1u);
        }
    }
    __syncthreads();
    suffix_select(hist, bsum, ssum, scal, KSEL);
    const uint32_t hi = scal[1];   // high-16 prefix holding the 127th key
    const uint32_t C1 = scal[2];   // #keys with prefix > hi
    __syncthreads();

    // ---- pass 2: histogram of key&0xFFFF within prefix hi (L2-resident) ----
    {
        uint4* h4 = (uint4*)hist;
        const uint4 z = make_uint4(0u, 0u, 0u, 0u);
        for (int i = tid; i < HIST_BINS / 4; i += TPB) h4[i] = z;
    }
    __syncthreads();
    const float4* l4 = (const float4*)lrow;
    for (int i = tid; i < NQ / 4; i += TPB) {
        float4 v = l4[i];
        #pragma unroll
        for (int e = 0; e < 4; ++e) {
            uint32_t idx = 4u * i + (uint32_t)e;
            if (idx != p) {
                uint32_t key = f2key((&v.x)[e]);
                if ((key >> 16) == hi) atomicAdd(&hist[key & 0xFFFFu], 1u);
            }
        }
    }
    __syncthreads();
    suffix_select(hist, bsum, ssum, scal, (uint32_t)KSEL - C1);
    const uint32_t lo = scal[1];
    const uint32_t G  = C1 + scal[2];           // #keys strictly greater than T
    const uint32_t E  = (uint32_t)KSEL - G;     // #keys == T to take (>=1)
    const uint32_t T  = (hi << 16) | lo;        // exact 127th-largest key
    __syncthreads();
    if (tid == 0) { scal[3] = 0u; scal[4] = 0u; }
    __syncthreads();

    // ---- pass 3: compact survivors (tie order is output-invariant) ----
    for (int i = tid; i < NQ / 4; i += TPB) {
        float4 v = l4[i];
        #pragma unroll
        for (int e = 0; e < 4; ++e) {
            uint32_t idx = 4u * i + (uint32_t)e;
            if (idx == p) continue;
            float f = (&v.x)[e];
            uint32_t key = f2key(f);
            if (key > T) {
                uint32_t pos = atomicAdd(&scal[3], 1u);
                if (pos < (uint32_t)KSEL) cand[pos] = f;     // exactly G fills
            } else if (key == T) {
                uint32_t eq = atomicAdd(&scal[4], 1u);
                if (eq < E) cand[G + eq] = f;                // exactly E fills
            }
        }
    }
    if (tid == 0) cand[KSEL] = -__builtin_inff();            // pad slot 127
    __syncthreads();

    // ---- 128-element bitonic sort in LDS, descending ----
    for (int k = 2; k <= KOUT; k <<= 1) {
        for (int j = k >> 1; j > 0; j >>= 1) {
            if (tid < KOUT) {
                int ixj = tid ^ j;
                if (ixj > tid) {
                    bool down = ((tid & k) == 0);            // descending segment
                    float a = cand[tid], b = cand[ixj];
                    if (down ? (a < b) : (a > b)) { cand[tid] = b; cand[ixj] = a; }
                }
            }
            __syncthreads();
        }
    }

    // ---- write outputs: slot 0 = positive, slots 1..127 = sorted negatives ----
    float* olog = out_logits + (size_t)row * KOUT;
    float* olab = out_labels + (size_t)row * KOUT;
    if (tid == 0) { olog[0] = lrow[p]; olab[0] = 1.0f; }
    if (tid >= 1 && tid < KOUT) { olog[tid] = cand[tid - 1]; olab[tid] = 0.0f; }
}

extern "C" void kernel_launch(void* const* d_in, const int* in_sizes, int n_in,
                              void* d_out, int out_size, void* d_ws, size_t ws_size,
                              hipStream_t stream) {
    (void)in_sizes; (void)n_in; (void)d_ws; (void)ws_size; (void)out_size;
    const float* logits = (const float*)d_in[0];
    const float* labels = (const float*)d_in[1];
    // d_in[2] = num_hard_negatives scalar (127) — baked into constants above.
    float* out_logits = (float*)d_out;
    float* out_labels = (float*)d_out + (size_t)BQ * KOUT;

    // allow 273 KB dynamic LDS (CDNA5 WGP supports up to 320 KB per workgroup)
    (void)hipFuncSetAttribute((const void*)hnm_topk_kernel,
                              hipFuncAttributeMaxDynamicSharedMemorySize,
                              SMEM_BYTES);
    hnm_topk_kernel<<<BQ, TPB, SMEM_BYTES, stream>>>(logits, labels,
                                                     out_logits, out_labels);
}